// GIN_51170240364736
// MI455X (gfx1250) — compile-verified
//
#include <hip/hip_runtime.h>
#include <hip/hip_bf16.h>

typedef float v2f __attribute__((ext_vector_type(2)));
typedef float v8f __attribute__((ext_vector_type(8)));

#define HID 128
#define LDS_PITCH 132   // 132 mod 64 banks = 4 -> conflict-free column reads

// Y[m, n] = sum_k X[m,k] * W[k,n] + b[n]
// Block = 256 threads = 8 waves. Block handles 16 rows x 128 cols.
// Wave w computes the 16x16 tile at columns [16w, 16w+16) using
// V_WMMA_F32_16X16X4_F32, accumulating over K=128 in steps of 4.
__global__ __launch_bounds__(256)
void gin_gemm(const float* __restrict__ X, const float* __restrict__ W,
              const float* __restrict__ bias, float* __restrict__ Y, int n_rows)
{
    __shared__ float sA[16 * LDS_PITCH];

    const int m0  = blockIdx.x * 16;
    const int tid = threadIdx.x;

    // Cooperative, coalesced load of the 16x128 A slab (float4 granularity).
    for (int i = tid; i < 16 * 32; i += 256) {
        const int r = i >> 5;       // row 0..15
        const int c = i & 31;       // float4 col 0..31
        float4 v = make_float4(0.f, 0.f, 0.f, 0.f);
        if (m0 + r < n_rows) {
            const float4* src =
                reinterpret_cast<const float4*>(X + (size_t)(m0 + r) * HID);
            v = src[c];
        }
        float* dst = &sA[r * LDS_PITCH + c * 4];
        dst[0] = v.x; dst[1] = v.y; dst[2] = v.z; dst[3] = v.w;
    }
    __syncthreads();

    const int wave = tid >> 5;           // 0..7 -> output column tile
    const int lane = tid & 31;
    const int row  = lane & 15;          // A-fragment row (M)
    const int kk   = (lane >> 4) * 2;    // A/B fragment K sub-offset (0 or 2)
    const int col  = wave * 16 + (lane & 15);  // B/D column (N)

    v8f acc = {};
#pragma unroll
    for (int k0 = 0; k0 < HID; k0 += 4) {
        // A 16x4 f32 fragment: lanes 0-15 hold K=k0..k0+1, lanes 16-31 K=k0+2..k0+3
        v2f a;
        a.x = sA[row * LDS_PITCH + k0 + kk];
        a.y = sA[row * LDS_PITCH + k0 + kk + 1];
        // B 4x16 f32 fragment (row-major W[k][n]): mirrored layout across columns
        v2f b;
        b.x = W[(size_t)(k0 + kk) * HID + col];
        b.y = W[(size_t)(k0 + kk + 1) * HID + col];
        acc = __builtin_amdgcn_wmma_f32_16x16x4_f32(
            /*neg_a=*/false, a, /*neg_b=*/false, b,
            /*c_mod=*/(short)0, acc, /*reuse_a=*/false, /*reuse_b=*/false);
    }

    // D layout: VGPR i -> M = i (lanes 0-15) / M = 8+i (lanes 16-31), N = lane&15
    const float bcol = bias[col];
    const int half = (lane >> 4) * 8;
#pragma unroll
    for (int i = 0; i < 8; ++i) {
        const int m = m0 + half + i;
        if (m < n_rows) Y[(size_t)m * HID + col] = acc[i] + bcol;
    }
}

// agg[rec[e], :] += h[send[e], :]   (agg pre-initialized with h, giving h+agg)
// One wave per edge: 32 lanes x float4 = coalesced 512B gather, then 4 f32
// atomics per lane into the L2-resident agg buffer.
__global__ __launch_bounds__(256)
void gin_scatter(const float* __restrict__ h, const int* __restrict__ send,
                 const int* __restrict__ rec, float* __restrict__ agg, int n_edges)
{
    const long long t = (long long)blockIdx.x * blockDim.x + threadIdx.x;
    const int e = (int)(t >> 5);
    if (e >= n_edges) return;
    const int lane = (int)(t & 31);

    const int s = send[e];
    const int r = rec[e];

    const float4* src = reinterpret_cast<const float4*>(h + (size_t)s * HID);
    const float4 v = src[lane];

    float* dst = agg + (size_t)r * HID + lane * 4;
    atomicAdd(dst + 0, v.x);
    atomicAdd(dst + 1, v.y);
    atomicAdd(dst + 2, v.z);
    atomicAdd(dst + 3, v.w);
}

extern "C" void kernel_launch(void* const* d_in, const int* in_sizes, int n_in,
                              void* d_out, int out_size, void* d_ws, size_t ws_size,
                              hipStream_t stream) {
    const float* h_in  = (const float*)d_in[0];
    const int*   eidx  = (const int*)d_in[1];
    const float* W_emb = (const float*)d_in[2];
    const float* b_emb = (const float*)d_in[3];
    const float* Ws    = (const float*)d_in[4];
    const float* bs    = (const float*)d_in[5];

    const int n_nodes = in_sizes[0] / HID;   // 100,000
    const int n_edges = in_sizes[1] / 2;     // 1,600,000
    const int* send = eidx;
    const int* rec  = eidx + n_edges;

    float* hbuf = (float*)d_out;             // node features live in d_out
    float* agg  = (float*)d_ws;              // 51.2 MB scratch (L2-resident)
    const size_t nh_bytes = (size_t)n_nodes * HID * sizeof(float);

    const int gemm_grid = (n_nodes + 15) / 16;
    const int scat_grid = (int)(((long long)n_edges * 32 + 255) / 256);

    // Input embedding: h = h_in @ W_embed + b_embed
    gin_gemm<<<gemm_grid, 256, 0, stream>>>(h_in, W_emb, b_emb, hbuf, n_nodes);

    for (int l = 0; l < 4; ++l) {
        // agg = h  (so scatter produces h + sum_{e: rec=v} h[send])
        hipMemcpyAsync(agg, hbuf, nh_bytes, hipMemcpyDeviceToDevice, stream);
        gin_scatter<<<scat_grid, 256, 0, stream>>>(hbuf, send, rec, agg, n_edges);
        // h = agg @ W_l + b_l
        gin_gemm<<<gemm_grid, 256, 0, stream>>>(
            agg, Ws + (size_t)l * HID * HID, bs + (size_t)l * HID, hbuf, n_nodes);
    }
}